// SpatialTransformerNetwork_24696061952319
// MI455X (gfx1250) — compile-verified
//
#include <hip/hip_runtime.h>
#include <hip/hip_bf16.h>
#include <math.h>

// Problem constants (from reference)
#define BB 16
#define CC 3
#define HH 256
#define WW 256
#define MM 100          // major_resolution
#define NTAP 121        // 11x11 blur taps
#define NSLICE 5        // output-row slices per (b,c) -> 16*3*5 = 240 workgroups
#define ROWS (MM / NSLICE)
#define THREADS 1024

typedef unsigned int u32x4 __attribute__((ext_vector_type(4)));
typedef int          i32x8 __attribute__((ext_vector_type(8)));
typedef int          i32x4 __attribute__((ext_vector_type(4)));

#ifndef __has_builtin
#define __has_builtin(x) 0
#endif

__global__ __launch_bounds__(THREADS)
void stn_blur_sample_kernel(const float* __restrict__ x,
                            const float* __restrict__ theta,
                            float* __restrict__ out)
{
    // 256 KB dynamic LDS: one full f32 channel image (256x256)
    extern __shared__ float lds_img[];

    // Packed tap table: {shift_x(px), shift_y(px), weight, pad} -> 1 ds_load_b128/tap
    __shared__ float4 s_tap[NTAP];
    __shared__ float  s_inv;

    const int slice = blockIdx.x;
    const int c     = blockIdx.y;
    const int b     = blockIdx.z;
    const int tid   = threadIdx.x;

    const float* gimg = x + (size_t)(b * CC + c) * (HH * WW);

    // ---- Stage the whole channel into LDS via the Tensor Data Mover ----
    // D#: 2D tensor, 4-byte elements, dim0=256 (stride 256), dim1=256,
    // tile = entire tensor. One DMA per workgroup, issued by wave 0 and
    // overlapped with the tap-table setup below.
#if __has_builtin(__builtin_amdgcn_tensor_load_to_lds)
    if (tid < 32) {   // wave 0 only; TDM instruction issues once per wave
        unsigned long long ga = (unsigned long long)(const void*)gimg;
        unsigned int lds_off  = (unsigned int)(unsigned long long)(const void*)&lds_img[0];
        u32x4 g0;
        g0[0] = 1u;                                   // count=1 (valid user D#)
        g0[1] = lds_off;                              // lds_addr (bytes)
        g0[2] = (unsigned int)ga;                     // global_addr[31:0]
        g0[3] = ((unsigned int)(ga >> 32) & 0x01FFFFFFu)
                | 0x80000000u;                        // global_addr[56:32] | type=2
        i32x8 g1;
        g1[0] = 0x00020000;   // data_size = 2 (4 bytes); no multicast/iterate/pad
        g1[1] = 0x01000000;   // tensor_dim0 = 256 (low16 @ bits 63:48)
        g1[2] = 0x01000000;   // tensor_dim1 = 256 (low16 @ bits 95:80)
        g1[3] = 0x01000000;   // tile_dim0   = 256 (bits 127:112)
        g1[4] = 0x00000100;   // tile_dim1   = 256 (bits 143:128), tile_dim2 = 0
        g1[5] = 0x00000100;   // tensor_dim0_stride = 256 (bits 191:160)
        g1[6] = 0x00000000;   // dim0_stride hi16 = 0, dim1_stride lo16 = 0
        g1[7] = 0x00000001;   // tensor_dim1_stride = 65536 (bits 255:224)
        i32x4 gz4 = {0, 0, 0, 0};
        i32x8 gz8 = {0, 0, 0, 0, 0, 0, 0, 0};
        // 6-arg form (clang-23 / therock-10.0):
        // (group0, group1, group2, group3, extra group, cpol)
        __builtin_amdgcn_tensor_load_to_lds(g0, g1, gz4, gz4, gz8, 0);
    }
#else
    for (int i = tid; i < (HH * WW) / 4; i += THREADS)
        ((float4*)lds_img)[i] = ((const float4*)gimg)[i];
#endif

    // ---- Per-batch affine + tap table (overlaps the DMA) ----
    const float t00 = theta[b*6+0], t01 = theta[b*6+1], t02 = theta[b*6+2];
    const float t10 = theta[b*6+3], t11 = theta[b*6+4], t12 = theta[b*6+5];

    if (tid < NTAP) {
        int i = tid / 11, j = tid % 11;
        // linspace(-1,1,13)[1:-1] : -1 + (k+1)/6
        float dx = -1.0f + (float)(i + 1) * (1.0f / 6.0f);
        float dy = -1.0f + (float)(j + 1) * (1.0f / 6.0f);
        float wu = __expf(-8.0f * (dx * dx + dy * dy));   // exp(-0.5*r2*16)
        // tap shift in input-pixel units: (theta_lin . d) * scale * (W/2)
        const float k = 128.0f / 99.0f;                   // (1/(M-1)) * 128
        s_tap[tid] = make_float4((t00 * dx + t01 * dy) * k,
                                 (t10 * dx + t11 * dy) * k,
                                 wu, 0.0f);
    }
    __syncthreads();
    if (tid == 0) {
        float s = 0.0f;
        for (int t = 0; t < NTAP; ++t) s += s_tap[t].z;
        s_inv = 1.0f / s;
    }
    __syncthreads();
    if (tid < NTAP) s_tap[tid].z *= s_inv;

#if __has_builtin(__builtin_amdgcn_tensor_load_to_lds)
    if (tid < 32) {
#if __has_builtin(__builtin_amdgcn_s_wait_tensorcnt)
        __builtin_amdgcn_s_wait_tensorcnt(0);
#else
        asm volatile("s_wait_tensorcnt 0" ::: "memory");
#endif
    }
#endif
    __syncthreads();   // image + tap table now visible to all waves

    // ---- 121-tap blurred bilinear sampling, all gathers from LDS ----
    for (int p = tid; p < ROWS * MM; p += THREADS) {
        const int h = slice * ROWS + p / MM;
        const int w = p % MM;
        // affine_grid base coords (align_corners=False pixel centers)
        const float xc = ((float)w + 0.5f) * (2.0f / MM) - 1.0f;
        const float yc = ((float)h + 0.5f) * (2.0f / MM) - 1.0f;
        // unnormalized base sample position: ((g+1)*256 - 1)/2
        const float bix = (t00 * xc + t01 * yc + t02 + 1.0f) * 128.0f - 0.5f;
        const float biy = (t10 * xc + t11 * yc + t12 + 1.0f) * 128.0f - 0.5f;

        float acc = 0.0f;
        for (int t = 0; t < NTAP; ++t) {
            const float4 tp = s_tap[t];        // one ds_load_b128
            const float ix = bix + tp.x;
            const float iy = biy + tp.y;
            const float x0 = floorf(ix), y0 = floorf(iy);
            const float fx = ix - x0,    fy = iy - y0;
            const float x1 = x0 + 1.0f,  y1 = y0 + 1.0f;
            // zero-padding: fold per-corner validity into the weights
            const float wx0 = (x0 >= 0.0f && x0 <= 255.0f) ? (1.0f - fx) : 0.0f;
            const float wx1 = (x1 >= 0.0f && x1 <= 255.0f) ? fx          : 0.0f;
            const float wy0 = (y0 >= 0.0f && y0 <= 255.0f) ? (1.0f - fy) : 0.0f;
            const float wy1 = (y1 >= 0.0f && y1 <= 255.0f) ? fy          : 0.0f;
            const int ix0 = (int)fminf(fmaxf(x0, 0.0f), 255.0f);
            const int ix1 = (int)fminf(fmaxf(x1, 0.0f), 255.0f);
            const int iy0 = (int)fminf(fmaxf(y0, 0.0f), 255.0f);
            const int iy1 = (int)fminf(fmaxf(y1, 0.0f), 255.0f);
            const float* r0 = lds_img + (iy0 << 8);
            const float* r1 = lds_img + (iy1 << 8);
            const float v = wy0 * (wx0 * r0[ix0] + wx1 * r0[ix1])
                          + wy1 * (wx0 * r1[ix0] + wx1 * r1[ix1]);
            acc += tp.z * v;
        }
        out[((size_t)(b * CC + c) * MM + h) * MM + w] = acc;
    }
}

extern "C" void kernel_launch(void* const* d_in, const int* in_sizes, int n_in,
                              void* d_out, int out_size, void* d_ws, size_t ws_size,
                              hipStream_t stream) {
    const float* x     = (const float*)d_in[0];   // [16,3,256,256] f32
    const float* theta = (const float*)d_in[1];   // [16,2,3] f32
    float* out = (float*)d_out;                   // [16,3,100,100] f32

    const size_t shmem = (size_t)HH * WW * sizeof(float);  // 256 KB dynamic LDS
    (void)hipFuncSetAttribute(reinterpret_cast<const void*>(stn_blur_sample_kernel),
                              hipFuncAttributeMaxDynamicSharedMemorySize,
                              (int)shmem);

    dim3 grid(NSLICE, CC, BB);   // 240 workgroups
    stn_blur_sample_kernel<<<grid, dim3(THREADS), shmem, stream>>>(x, theta, out);
}